// MedianBlur_25262997635417
// MI455X (gfx1250) — compile-verified
//
#include <hip/hip_runtime.h>
#include <cstdint>

// 5x5 median blur, zero padding, lower median (== true median for 25 elems).
// Input : (4,3,1024,1024) f32  -> 12 planes of 1024x1024
// Output: same shape f32
//
// Staging: CDNA5 async DMA (global_load_async_to_lds_b128, ASYNCcnt) into a
// 72x20 LDS tile (aligned 16B chunks; OOB chunks zero-filled with ds_store).
// Compute: VALU-bound 99-compare-exchange median network, 4 outputs/thread.

#define IMG_H 1024
#define IMG_W 1024
#define TILE_W 64
#define TILE_H 16
#define LDS_W 72              // TILE_W + 8 (aligned halo: [tileX0-4, tileX0+68))
#define LDS_H (TILE_H + 4)    // 20
#define LDS_CHUNKS 18         // LDS_W / 4 float4 chunks per row
#define N_CELLS (LDS_H * LDS_CHUNKS)   // 360

__device__ __forceinline__ void cswap(float& a, float& b) {
    float mn = __builtin_fminf(a, b);
    b = __builtin_fmaxf(a, b);
    a = mn;
}

// 99-compare-exchange median-of-25 network (Smith / McGuire). Median -> v[12].
__device__ __forceinline__ float median25(float v[25]) {
#define CX(i, j) cswap(v[i], v[j])
    CX(0,1);  CX(3,4);  CX(2,4);  CX(2,3);  CX(6,7);
    CX(5,7);  CX(5,6);  CX(9,10); CX(8,10); CX(8,9);
    CX(12,13);CX(11,13);CX(11,12);CX(15,16);CX(14,16);
    CX(14,15);CX(18,19);CX(17,19);CX(17,18);CX(21,22);
    CX(20,22);CX(20,21);CX(23,24);CX(2,5);  CX(3,6);
    CX(0,6);  CX(0,3);  CX(4,7);  CX(1,7);  CX(1,4);
    CX(11,14);CX(8,14); CX(8,11); CX(12,15);CX(9,15);
    CX(9,12); CX(13,16);CX(10,16);CX(10,13);CX(20,23);
    CX(17,23);CX(17,20);CX(21,24);CX(18,24);CX(18,21);
    CX(19,22);CX(8,17); CX(9,18); CX(0,18); CX(0,9);
    CX(10,19);CX(1,19); CX(1,10); CX(11,20);CX(2,20);
    CX(2,11); CX(12,21);CX(3,21); CX(3,12); CX(13,22);
    CX(4,22); CX(4,13); CX(14,23);CX(5,23); CX(5,14);
    CX(15,24);CX(6,24); CX(6,15); CX(7,16); CX(7,19);
    CX(13,21);CX(15,23);CX(7,13); CX(7,15); CX(1,9);
    CX(3,11); CX(5,17); CX(11,17);CX(9,17); CX(4,10);
    CX(6,12); CX(7,14); CX(4,6);  CX(4,7);  CX(12,14);
    CX(10,14);CX(6,7);  CX(10,12);CX(6,10); CX(6,17);
    CX(12,17);CX(7,17); CX(7,10); CX(12,18);CX(7,12);
    CX(10,18);CX(12,20);CX(10,20);CX(10,12);
#undef CX
    return v[12];
}

__global__ __launch_bounds__(256) void median5x5_kernel(const float* __restrict__ x,
                                                        float* __restrict__ out) {
    __shared__ __align__(16) float tile[LDS_H * LDS_W];

    const int tid    = threadIdx.x;
    const int plane  = blockIdx.z;
    const int tileX0 = blockIdx.x * TILE_W;   // multiple of 64 -> 16B-aligned chunks
    const int tileY0 = blockIdx.y * TILE_H;

    const float* src = x + (size_t)plane * (IMG_H * IMG_W);

    // Low 32 bits of a generic pointer into LDS == byte offset within LDS
    // (flat->LDS aperture truncation), which is what the async op's LDS VGPR wants.
    const uint32_t ldsBase = (uint32_t)(uintptr_t)&tile[0];

    // --- Stage halo tile into LDS with CDNA5 async DMA, 16B chunks. ---
    // Staged x-range [tileX0-4, tileX0+68) is chunk-aligned, and since IMG_W is a
    // multiple of 4 every chunk is entirely in-bounds or entirely OOB in x.
    // In-bounds chunk : global_load_async_to_lds_b128  (ASYNCcnt path)
    // OOB chunk       : ds_store_b128 of zeros          (disjoint addresses)
#pragma unroll
    for (int it = 0; it < 2; ++it) {
        const int i = tid + it * 256;
        if (i < N_CELLS) {
            const int ly = i / LDS_CHUNKS;
            const int c  = i - ly * LDS_CHUNKS;
            const int gy = tileY0 + ly - 2;
            const int gx = tileX0 + c * 4 - 4;
            if ((unsigned)gy < (unsigned)IMG_H && (unsigned)gx < (unsigned)IMG_W) {
                const uint64_t gaddr = (uint64_t)(uintptr_t)(src + (size_t)gy * IMG_W + gx);
                const uint32_t laddr = ldsBase + (uint32_t)(ly * (LDS_W * 4) + c * 16);
                asm volatile("global_load_async_to_lds_b128 %0, %1, off"
                             :
                             : "v"(laddr), "v"(gaddr)
                             : "memory");
            } else {
                float4 z = make_float4(0.f, 0.f, 0.f, 0.f);
                *(float4*)&tile[ly * LDS_W + c * 4] = z;   // ds_store_b128
            }
        }
    }
    // Wait for this wave's async LDS writes; barrier covers the DS zero-fill
    // (compiler inserts dscnt waits) and makes all waves' tiles visible.
    asm volatile("s_wait_asynccnt 0x0" ::: "memory");
    __syncthreads();

    // --- Compute: each thread does a 1-wide x 4-tall strip of outputs. ---
    const int tx = tid & (TILE_W - 1);   // 0..63 output column in tile
    const int tq = tid >> 6;             // 0..3
    const int y0 = tq * 4;               // first output row (local) of strip

    // 8 rows x 5 cols register window covers all four 5x5 windows of the strip:
    // 40 LDS reads for 4 outputs instead of 100. Window cols = tx+2 .. tx+6
    // (staged range starts 4 left of the tile; window needs 2 left of output col).
    float wbuf[8][5];
#pragma unroll
    for (int r = 0; r < 8; ++r) {
#pragma unroll
        for (int c = 0; c < 5; ++c) {
            wbuf[r][c] = tile[(y0 + r) * LDS_W + tx + 2 + c];
        }
    }

    float* dst = out + (size_t)plane * (IMG_H * IMG_W)
                     + (size_t)(tileY0 + y0) * IMG_W + (tileX0 + tx);
#pragma unroll
    for (int r = 0; r < 4; ++r) {
        float v[25];
#pragma unroll
        for (int dy = 0; dy < 5; ++dy) {
#pragma unroll
            for (int dx = 0; dx < 5; ++dx) {
                v[dy * 5 + dx] = wbuf[r + dy][dx];
            }
        }
        dst[(size_t)r * IMG_W] = median25(v);
    }
}

extern "C" void kernel_launch(void* const* d_in, const int* in_sizes, int n_in,
                              void* d_out, int out_size, void* d_ws, size_t ws_size,
                              hipStream_t stream) {
    const float* x = (const float*)d_in[0];
    float* out = (float*)d_out;
    const int planes = in_sizes[0] / (IMG_H * IMG_W);   // 4*3 = 12
    dim3 grid(IMG_W / TILE_W, IMG_H / TILE_H, planes);  // (16, 64, 12)
    median5x5_kernel<<<grid, dim3(256, 1, 1), 0, stream>>>(x, out);
}